// X_conv_64037962384011
// MI455X (gfx1250) — compile-verified
//
#include <hip/hip_runtime.h>
#include <hip/hip_bf16.h>

typedef __attribute__((ext_vector_type(2))) float v2f;
typedef __attribute__((ext_vector_type(8))) float v8f;

#define P_TOTAL 32768   // B*N
#define KN 16
#define CIN 160
#define EPSF 1e-5f

static __device__ __forceinline__ v8f wmma4(v2f a, v2f b, v8f c) {
    // V_WMMA_F32_16X16X4_F32 : D = A(16x4) * B(4x16) + C
    return __builtin_amdgcn_wmma_f32_16x16x4_f32(false, a, false, b, (short)0, c, false, false);
}

// ---------------------------------------------------------------- utilities
__global__ void k_zero(float* p, int n) {
    int t = blockIdx.x * 256 + threadIdx.x;
    if (t < n) p[t] = 0.f;
}

// Transpose weights to K-major AND interleave even/odd K-row pairs so a WMMA
// B-fragment (rows k,k+1 at column n) is one aligned b64 load:
//   wt[((k>>1)*256 + n)*2 + (k&1)] = W[k][n]
__global__ void k_prep(const float* __restrict__ W1, const float* __restrict__ W2,
                       const float* __restrict__ W3, const float* __restrict__ Wc,
                       float* __restrict__ w1t, float* __restrict__ w2t,
                       float* __restrict__ w3t, float* __restrict__ wct) {
    int t = blockIdx.x * 256 + threadIdx.x;
    if (t < 48 * 256) {
        int k = t >> 8, n = t & 255;
        w1t[(((k >> 1) * 256 + n) << 1) + (k & 1)] = W1[n * 48 + k];
        return;
    }
    int t2 = t - 48 * 256;
    if (t2 < 65536) {
        int k = t2 >> 8, n = t2 & 255;
        w2t[(((k >> 1) * 256 + n) << 1) + (k & 1)] = W2[n * 256 + k];
        return;
    }
    int t3 = t2 - 65536;
    if (t3 < 65536) {
        int k = t3 >> 8, n = t3 & 255;
        w3t[(((k >> 1) * 256 + n) << 1) + (k & 1)] = W3[n * 256 + k];
        return;
    }
    int t4 = t3 - 65536;
    if (t4 < 2560 * 256) {   // q = kk*160+c ; Wc[o][c][kk]
        int q = t4 >> 8, o = t4 & 255;
        int kk = q / 160, c = q % 160;
        wct[(((q >> 1) * 256 + o) << 1) + (q & 1)] = Wc[o * 2560 + c * 16 + kk];
    }
}

// pts = points - reps (stored (P,48) row-major); lift channel sum/sumsq; reps passthrough
__global__ void k_pts_lift(const float* __restrict__ points, const float* __restrict__ reps,
                           const float* __restrict__ Wl, const float* __restrict__ bl,
                           float* __restrict__ ptsb, float* __restrict__ liftSum,
                           float* __restrict__ liftSq, float* __restrict__ repOut) {
    int gid = blockIdx.x * 256 + threadIdx.x;   // one (p,k) pair
    int p = gid >> 4, k = gid & 15;
    int lane = threadIdx.x & 31;
    float rx = reps[p * 3 + 0], ry = reps[p * 3 + 1], rz = reps[p * 3 + 2];
    float x = points[gid * 3 + 0] - rx;
    float y = points[gid * 3 + 1] - ry;
    float z = points[gid * 3 + 2] - rz;
    ptsb[gid * 3 + 0] = x;   // == p*48 + k*3
    ptsb[gid * 3 + 1] = y;
    ptsb[gid * 3 + 2] = z;
    if (k == 0) {
        repOut[p * 3 + 0] = rx;
        repOut[p * 3 + 1] = ry;
        repOut[p * 3 + 2] = rz;
    }
    __shared__ float sS[32], sQ[32];
    if (threadIdx.x < 32) { sS[threadIdx.x] = 0.f; sQ[threadIdx.x] = 0.f; }
    __syncthreads();
    for (int l = 0; l < 32; ++l) {
        float v = fmaf(x, Wl[l * 3 + 0], fmaf(y, Wl[l * 3 + 1], fmaf(z, Wl[l * 3 + 2], bl[l])));
        float s = v, q = v * v;
        for (int off = 16; off; off >>= 1) {
            s += __shfl_xor(s, off, 32);
            q += __shfl_xor(q, off, 32);
        }
        if (lane == 0) { atomicAdd(&sS[l], s); atomicAdd(&sQ[l], q); }
    }
    __syncthreads();
    if (threadIdx.x < 32) {
        atomicAdd(&liftSum[threadIdx.x], sS[threadIdx.x]);
        atomicAdd(&liftSq[threadIdx.x], sQ[threadIdx.x]);
    }
}

// finalize BN: scale = g*rsqrt(var+eps); shift = beta - mean*scale
__global__ void k_bnfin(const float* __restrict__ sum, const float* __restrict__ sq,
                        const float* __restrict__ gamma, const float* __restrict__ beta,
                        float* __restrict__ scale, float* __restrict__ shift,
                        int C, float invCnt) {
    int c = threadIdx.x + blockIdx.x * blockDim.x;
    if (c >= C) return;
    float m = sum[c] * invCnt;
    float v = sq[c] * invCnt - m * m;
    float inv = rsqrtf(v + EPSF);
    float s = gamma[c] * inv;
    scale[c] = s;
    shift[c] = beta[c] - m * s;
}

// per-channel sum/sumsq over a (P,256) matrix, coalesced, 128 rows per block
__global__ void k_colstats(const float* __restrict__ X, float* __restrict__ sum,
                           float* __restrict__ sq) {
    int c = threadIdx.x;
    int r0 = blockIdx.x * 128;
    float s = 0.f, q = 0.f;
    for (int r = 0; r < 128; ++r) {
        float x = X[(r0 + r) * 256 + c];
        s += x;
        q += x * x;
    }
    atomicAdd(&sum[c], s);
    atomicAdd(&sq[c], q);
}

// WMMA GEMM: C(P,256) = act(A(P,KD)) @ W(KD,256) + bias
// TRANS: apply per-K-channel BN scale/shift + ReLU to the A operand.
// Wt is pair-interleaved K-major (see k_prep).
template <int KD, bool TRANS>
__global__ void k_gemm16(const float* __restrict__ A,
                         const float* __restrict__ Wt, const float* __restrict__ bias,
                         const float* __restrict__ sc, const float* __restrict__ sh,
                         float* __restrict__ Cout) {
    int lane = threadIdx.x & 31, w = threadIdx.x >> 5;
    int idx = lane & 15, half = lane >> 4;
    int r0 = blockIdx.x * 16;
    int n0a = 32 * w, n0b = 32 * w + 16;
    v8f acc0 = {}, acc1 = {};
    const float* Arow = A + (size_t)(r0 + idx) * KD;
    for (int k0 = 0; k0 < KD; k0 += 4) {
        int k = k0 + 2 * half;                      // even
        float x0 = Arow[k], x1 = Arow[k + 1];
        if (TRANS) {
            x0 = fmaxf(fmaf(x0, sc[k], sh[k]), 0.f);
            x1 = fmaxf(fmaf(x1, sc[k + 1], sh[k + 1]), 0.f);
        }
        v2f a; a.x = x0; a.y = x1;
        const float* wrow = Wt + (size_t)(k >> 1) * 512;
        v2f b0 = *(const v2f*)(wrow + ((n0a + idx) << 1));
        v2f b1 = *(const v2f*)(wrow + ((n0b + idx) << 1));
        acc0 = wmma4(a, b0, acc0);
        acc1 = wmma4(a, b1, acc1);
    }
#pragma unroll
    for (int v = 0; v < 8; ++v) {
        int r = r0 + v + 8 * half;
        Cout[(size_t)r * 256 + n0a + idx] = acc0[v] + bias[n0a + idx];
        Cout[(size_t)r * 256 + n0b + idx] = acc1[v] + bias[n0b + idx];
    }
}

// ---------------------------------------------------------------- fused final stage
// One 16-channel chunk: phase 1 builds fm chunk in LDS, phase 2 accumulates into out.
template <bool LIFT>
static __device__ __forceinline__ void do_chunk(
        int cc0, int w, int idx, int half, int p0, int o0a, int o0b,
        const v2f tf[2][4], float* __restrict__ fmC,
        const float* __restrict__ ptsb, const float* __restrict__ Wl,
        const float* __restrict__ bl, const float* __restrict__ lsc,
        const float* __restrict__ lsh, const float* __restrict__ features,
        const float* __restrict__ wct, v8f& accA, v8f& accB) {
    // phase 1: this wave computes the fm chunk of its 2 points
    for (int q = 0; q < 2; ++q) {
        int p = p0 + 2 * w + q;
        v8f f = {};
#pragma unroll
        for (int s = 0; s < 4; ++s) {
            int j0 = s * 4 + 2 * half;
            v2f b;
            if (LIFT) {                              // recompute lift + BN + ReLU
                int l = cc0 + idx;
                float w0 = Wl[l * 3], w1 = Wl[l * 3 + 1], w2 = Wl[l * 3 + 2];
                float bb = bl[l], s_ = lsc[l], h_ = lsh[l];
                const float* pr = ptsb + (size_t)p * 48 + j0 * 3;
                float v0 = fmaf(pr[0], w0, fmaf(pr[1], w1, fmaf(pr[2], w2, bb)));
                float v1 = fmaf(pr[3], w0, fmaf(pr[4], w1, fmaf(pr[5], w2, bb)));
                b.x = fmaxf(fmaf(v0, s_, h_), 0.f);
                b.y = fmaxf(fmaf(v1, s_, h_), 0.f);
            } else {                                 // raw input features
                int c = cc0 - 32 + idx;
                b.x = features[(size_t)p * 2048 + j0 * 128 + c];
                b.y = features[(size_t)p * 2048 + (j0 + 1) * 128 + c];
            }
            f = wmma4(tf[q][s], b, f);
        }
        int base = (2 * w + q) * 256;
#pragma unroll
        for (int v = 0; v < 8; ++v)
            fmC[base + (v + 8 * half) * 16 + idx] = f[v];
    }
    __syncthreads();
    // phase 2: out += fm_chunk @ WcT (wct pair-interleaved; Q always even)
    for (int s2 = 0; s2 < 64; ++s2) {
        int q0 = 4 * s2 + 2 * half;
        v2f a;
        a.x = fmC[idx * 256 + q0];
        a.y = fmC[idx * 256 + q0 + 1];
        int i1 = q0 >> 4, ccq = q0 & 15;
        int Qh = (i1 * 160 + cc0 + ccq) >> 1;
        const float* wr = wct + (size_t)Qh * 512;
        v2f b0 = *(const v2f*)(wr + ((o0a + idx) << 1));
        v2f b1 = *(const v2f*)(wr + ((o0b + idx) << 1));
        accA = wmma4(a, b0, accA);
        accB = wmma4(a, b1, accB);
    }
    __syncthreads();
}

__global__ void k_final(const float* __restrict__ t3raw, const float* __restrict__ sc3,
                        const float* __restrict__ sh3, const float* __restrict__ ptsb,
                        const float* __restrict__ Wl, const float* __restrict__ bl,
                        const float* __restrict__ lsc, const float* __restrict__ lsh,
                        const float* __restrict__ features, const float* __restrict__ wct,
                        const float* __restrict__ bc, float* __restrict__ outp) {
    __shared__ float fmC[16 * 256];   // 16 points x (16 i x 16 c) chunk of fm
    int lane = threadIdx.x & 31, w = threadIdx.x >> 5;
    int idx = lane & 15, half = lane >> 4;
    int p0 = blockIdx.x * 16;
    int o0a = 32 * w, o0b = 32 * w + 16;

    // preload T fragments for this wave's two points (BN3 applied)
    v2f tf[2][4];
    for (int q = 0; q < 2; ++q) {
        int p = p0 + 2 * w + q;
#pragma unroll
        for (int s = 0; s < 4; ++s) {
            int ch = idx * 16 + s * 4 + 2 * half;
            tf[q][s].x = fmaf(t3raw[(size_t)p * 256 + ch], sc3[ch], sh3[ch]);
            tf[q][s].y = fmaf(t3raw[(size_t)p * 256 + ch + 1], sc3[ch + 1], sh3[ch + 1]);
        }
    }

    v8f accA = {}, accB = {};
    // chunks 0..1: lift channels (recomputed); chunks 2..9: raw features
    for (int cc0 = 0; cc0 < 32; cc0 += 16)
        do_chunk<true>(cc0, w, idx, half, p0, o0a, o0b, tf, fmC,
                       ptsb, Wl, bl, lsc, lsh, features, wct, accA, accB);
    for (int cc0 = 32; cc0 < CIN; cc0 += 16)
        do_chunk<false>(cc0, w, idx, half, p0, o0a, o0b, tf, fmC,
                        ptsb, Wl, bl, lsc, lsh, features, wct, accA, accB);

#pragma unroll
    for (int v = 0; v < 8; ++v) {
        int p = p0 + v + 8 * half;
        outp[(size_t)p * 256 + o0a + idx] = accA[v] + bc[o0a + idx];
        outp[(size_t)p * 256 + o0b + idx] = accB[v] + bc[o0b + idx];
    }
}

// ---------------------------------------------------------------- launch
extern "C" void kernel_launch(void* const* d_in, const int* in_sizes, int n_in,
                              void* d_out, int out_size, void* d_ws, size_t ws_size,
                              hipStream_t stream) {
    const float* points   = (const float*)d_in[0];
    const float* features = (const float*)d_in[1];
    const float* reps     = (const float*)d_in[2];
    const float* Wl  = (const float*)d_in[3];
    const float* bl  = (const float*)d_in[4];
    const float* gl  = (const float*)d_in[5];
    const float* betl= (const float*)d_in[6];
    const float* W1 = (const float*)d_in[7];
    const float* b1 = (const float*)d_in[8];
    const float* g1 = (const float*)d_in[9];
    const float* be1= (const float*)d_in[10];
    const float* W2 = (const float*)d_in[11];
    const float* b2 = (const float*)d_in[12];
    const float* g2 = (const float*)d_in[13];
    const float* be2= (const float*)d_in[14];
    const float* W3 = (const float*)d_in[15];
    const float* b3 = (const float*)d_in[16];
    const float* g3 = (const float*)d_in[17];
    const float* be3= (const float*)d_in[18];
    const float* Wc = (const float*)d_in[19];
    const float* bc = (const float*)d_in[20];

    float* dout = (float*)d_out;
    float* out2 = dout + (size_t)P_TOTAL * 3;   // second output after reps passthrough

    // workspace layout (floats)
    float* ws    = (float*)d_ws;
    float* stats = ws;                               // 4096
    float* ptsb  = ws + 4096;                        // P*48
    float* w1t   = ptsb + (size_t)P_TOTAL * 48;      // 48*256
    float* w2t   = w1t + 48 * 256;                   // 256*256
    float* w3t   = w2t + 65536;                      // 256*256
    float* wct   = w3t + 65536;                      // 2560*256
    float* t1    = wct + 2560 * 256;                 // P*256
    float* t2    = t1 + (size_t)P_TOTAL * 256;       // P*256
    float* t3    = t1;                               // reuse t1 slot

    float* liftSum = stats + 0,   *liftSq = stats + 32;
    float* liftSc  = stats + 64,  *liftSh = stats + 96;
    float* s1Sum = stats + 128,  *s1Sq = stats + 384, *sc1 = stats + 640,  *sh1 = stats + 896;
    float* s2Sum = stats + 1152, *s2Sq = stats + 1408,*sc2 = stats + 1664, *sh2 = stats + 1920;
    float* s3Sum = stats + 2176, *s3Sq = stats + 2432,*sc3 = stats + 2688, *sh3 = stats + 2944;

    k_zero<<<16, 256, 0, stream>>>(stats, 4096);
    k_prep<<<3120, 256, 0, stream>>>(W1, W2, W3, Wc, w1t, w2t, w3t, wct);
    k_pts_lift<<<(P_TOTAL * KN) / 256, 256, 0, stream>>>(points, reps, Wl, bl,
                                                         ptsb, liftSum, liftSq, dout);
    k_bnfin<<<1, 32, 0, stream>>>(liftSum, liftSq, gl, betl, liftSc, liftSh,
                                  32, 1.f / (float)(P_TOTAL * KN));

    k_gemm16<48, false><<<P_TOTAL / 16, 256, 0, stream>>>(ptsb, w1t, b1, nullptr, nullptr, t1);
    k_colstats<<<P_TOTAL / 128, 256, 0, stream>>>(t1, s1Sum, s1Sq);
    k_bnfin<<<1, 256, 0, stream>>>(s1Sum, s1Sq, g1, be1, sc1, sh1, 256, 1.f / (float)P_TOTAL);

    k_gemm16<256, true><<<P_TOTAL / 16, 256, 0, stream>>>(t1, w2t, b2, sc1, sh1, t2);
    k_colstats<<<P_TOTAL / 128, 256, 0, stream>>>(t2, s2Sum, s2Sq);
    k_bnfin<<<1, 256, 0, stream>>>(s2Sum, s2Sq, g2, be2, sc2, sh2, 256, 1.f / (float)P_TOTAL);

    k_gemm16<256, true><<<P_TOTAL / 16, 256, 0, stream>>>(t2, w3t, b3, sc2, sh2, t3);
    k_colstats<<<P_TOTAL / 128, 256, 0, stream>>>(t3, s3Sum, s3Sq);
    k_bnfin<<<1, 256, 0, stream>>>(s3Sum, s3Sq, g3, be3, sc3, sh3, 256, 1.f / (float)P_TOTAL);

    k_final<<<P_TOTAL / 16, 256, 0, stream>>>(t3, sc3, sh3, ptsb, Wl, bl, liftSc, liftSh,
                                              features, wct, bc, out2);
}